// GATModel_11244224381604
// MI455X (gfx1250) — compile-verified
//
#include <hip/hip_runtime.h>
#include <math.h>

// ---------------------------------------------------------------------------
// CDNA5 (gfx1250) GAT forward: 3x (GEMM -> attention softmax -> scatter).
// GEMMs use V_WMMA_F32_16X16X4_F32 (fp32-exact matrix path, wave32).
// B staged in LDS in pair-interleaved fragment layout -> one ds_load_b64
// per B fragment, zero register repacking before each WMMA.
// ---------------------------------------------------------------------------

typedef float v2f __attribute__((ext_vector_type(2)));
typedef float v8f __attribute__((ext_vector_type(8)));

#define NEG_SLOPE 0.2f

// ---------------------------------------------------------------------------
// WMMA fp32 GEMM: C[M x Nout] = A[M x K] * W[K x Nout]
//   - block = 128 threads = 4 waves; each wave computes a 16-row M tile
//   - NT column tiles of 16 (Nout zero-padded to NT*16 in LDS)
//   - A fragment (16x4 f32): lane m = lane&15; lanes>=16 hold K+2,K+3
//   - B staged as Bs[kpair][n][2] (K-adjacent pair contiguous) with row
//     stride padded to ===32 (mod 64) banks so the two lane halves (which
//     read kpair and kpair+1) touch disjoint bank halves.
//   - C/D layout per ISA: VGPR j -> row j (lanes 0-15) / row j+8 (lanes 16-31)
// Requires M % 16 == 0 and K % 4 == 0 (true here: 50000, 256/48).
// ---------------------------------------------------------------------------
template <int KMAX, int NT>
__global__ void __launch_bounds__(128)
wmma_gemm_f32(const float* __restrict__ A, const float* __restrict__ W,
              float* __restrict__ C, int M, int K, int Nout) {
  constexpr int NPAD = NT * 16;
  // row stride (floats) for one kpair row; force stride % 64 == 32
  constexpr int LSTRIDE = NPAD * 2 + ((32 - ((NPAD * 2) & 63)) & 63);
  __shared__ float Bs[(KMAX / 2) * LSTRIDE];

  const int tid = threadIdx.x;
  // Stage W (K x Nout) -> Bs[kpair * LSTRIDE + n*2 + (k&1)], zero-padded cols.
  for (int idx = tid; idx < K * NPAD; idx += 128) {
    const int k = idx / NPAD;
    const int n = idx - k * NPAD;
    Bs[(k >> 1) * LSTRIDE + n * 2 + (k & 1)] =
        (n < Nout) ? W[(size_t)k * Nout + n] : 0.0f;
  }
  __syncthreads();

  const int wave = tid >> 5;
  const int lane = tid & 31;
  const int mtile = blockIdx.x * 4 + wave;
  if (mtile * 16 >= M) return;  // wave-uniform exit: EXEC stays all-1s for WMMA

  const int row0 = mtile * 16;
  const int m = lane & 15;
  const int n = lane & 15;
  const int kb = (lane >> 4) << 1;  // 0 for lanes 0-15, 2 for lanes 16-31

  v8f acc[NT];
#pragma unroll
  for (int t = 0; t < NT; ++t) acc[t] = (v8f){};

  const float* __restrict__ arow = A + (size_t)(row0 + m) * K;
  // lane-invariant bases: fragment element pairs are contiguous & 8B aligned
  const v2f* __restrict__ afrag = (const v2f*)(arow + kb);        // index k/2
  const v2f* __restrict__ bbase = (const v2f*)(Bs + (kb >> 1) * LSTRIDE) + n;

#pragma unroll 4
  for (int k = 0; k < K; k += 4) {
    const v2f a = afrag[k >> 1];                       // global_load_b64
    const v2f* __restrict__ brow = bbase + (k >> 1) * (LSTRIDE / 2);
#pragma unroll
    for (int t = 0; t < NT; ++t) {
      const v2f b = brow[t * 16];                      // ds_load_b64
      acc[t] = __builtin_amdgcn_wmma_f32_16x16x4_f32(
          /*neg_a=*/false, a, /*neg_b=*/false, b,
          /*c_mod=*/(short)0, acc[t], /*reuse_a=*/false, /*reuse_b=*/false);
    }
  }

  const int rbase = (lane < 16) ? 0 : 8;
#pragma unroll
  for (int t = 0; t < NT; ++t) {
    const int col = t * 16 + n;
    if (col < Nout) {
#pragma unroll
      for (int j = 0; j < 8; ++j)
        C[(size_t)(row0 + rbase + j) * Nout + col] = acc[t][j];
    }
  }
}

// ---------------------------------------------------------------------------
// Elementwise helpers
// ---------------------------------------------------------------------------
__global__ void fill_f32(float* __restrict__ p, float v, long long n) {
  long long i = (long long)blockIdx.x * blockDim.x + threadIdx.x;
  if (i < n) p[i] = v;
}

// per-node attention coefficients: asrc[i] = h[i,:].a_src ; adst[i] = h[i,:].a_dst
__global__ void node_alpha(const float* __restrict__ h,
                           const float* __restrict__ aw_src,
                           const float* __restrict__ aw_dst,
                           float* __restrict__ asrc, float* __restrict__ adst,
                           int N, int F) {
  int i = blockIdx.x * blockDim.x + threadIdx.x;
  if (i >= N) return;
  const float* row = h + (size_t)i * F;
  float s1 = 0.0f, s2 = 0.0f;
  for (int f = 0; f < F; ++f) {
    const float v = row[f];
    s1 = fmaf(v, aw_src[f], s1);
    s2 = fmaf(v, aw_dst[f], s2);
  }
  asrc[i] = s1;
  adst[i] = s2;
}

__device__ __forceinline__ void atomic_max_f32(float* addr, float val) {
  // sign-split trick: correct for any mix of signs given -inf init
  if (val >= 0.0f)
    atomicMax((int*)addr, __float_as_int(val));
  else
    atomicMin((unsigned int*)addr, __float_as_uint(val));
}

__device__ __forceinline__ void edge_nodes(const long long* __restrict__ ei,
                                           long long E, long long e,
                                           int& s, int& d) {
  if (e < E) {
    s = (int)ei[e];
    d = (int)ei[E + e];
  } else {  // self loop
    s = d = (int)(e - E);
  }
}

// pass 1: leaky-relu logit per edge; segment max into mbuf[dst]
__global__ void edge_logit_max(const long long* __restrict__ ei, long long E,
                               long long ET, const float* __restrict__ asrc,
                               const float* __restrict__ adst,
                               float* __restrict__ elog,
                               float* __restrict__ mbuf) {
  long long e = (long long)blockIdx.x * blockDim.x + threadIdx.x;
  if (e >= ET) return;
  int s, d;
  edge_nodes(ei, E, e, s, d);
  float l = asrc[s] + adst[d];
  l = (l > 0.0f) ? l : NEG_SLOPE * l;
  elog[e] = l;
  atomic_max_f32(mbuf + d, l);
}

// mbuf = isfinite(mbuf) ? mbuf : 0   (reference semantics)
__global__ void sanitize_max(float* __restrict__ mbuf, int N) {
  int i = blockIdx.x * blockDim.x + threadIdx.x;
  if (i >= N) return;
  const float v = mbuf[i];
  if (isinf(v) || isnan(v)) mbuf[i] = 0.0f;
}

// pass 2: e = exp(logit - m[dst]); segment sum into sbuf[dst]
__global__ void edge_exp_sum(const long long* __restrict__ ei, long long E,
                             long long ET, const float* __restrict__ mbuf,
                             float* __restrict__ elog,
                             float* __restrict__ sbuf) {
  long long e = (long long)blockIdx.x * blockDim.x + threadIdx.x;
  if (e >= ET) return;
  int s, d;
  edge_nodes(ei, E, e, s, d);
  const float ev = __expf(elog[e] - mbuf[d]);
  elog[e] = ev;
  atomicAdd(sbuf + d, ev);
}

// pass 3: out[dst,:] += (e / (sum[dst]+eps)) * h[src,:]
// F % 4 == 0 path uses float4 gathers (h rows are L2-resident).
__global__ void edge_aggregate(const long long* __restrict__ ei, long long E,
                               long long ET, const float* __restrict__ elog,
                               const float* __restrict__ sbuf,
                               const float* __restrict__ h,
                               float* __restrict__ out, int F) {
  long long e = (long long)blockIdx.x * blockDim.x + threadIdx.x;
  if (e >= ET) return;
  int s, d;
  edge_nodes(ei, E, e, s, d);
  const float alpha = elog[e] / (sbuf[d] + 1e-16f);
  const float* __restrict__ hrow = h + (size_t)s * F;
  float* __restrict__ orow = out + (size_t)d * F;
  if ((F & 3) == 0) {
    const float4* __restrict__ h4 = (const float4*)hrow;
    for (int f4 = 0; f4 < (F >> 2); ++f4) {
      const float4 v = h4[f4];  // global_load_b128 gather
      atomicAdd(orow + f4 * 4 + 0, alpha * v.x);
      atomicAdd(orow + f4 * 4 + 1, alpha * v.y);
      atomicAdd(orow + f4 * 4 + 2, alpha * v.z);
      atomicAdd(orow + f4 * 4 + 3, alpha * v.w);
    }
  } else {
    for (int f = 0; f < F; ++f) atomicAdd(orow + f, alpha * hrow[f]);
  }
}

// out = (relu?) (out + b)
__global__ void bias_act(float* __restrict__ out, const float* __restrict__ b,
                         long long NF, int F, int do_relu) {
  long long i = (long long)blockIdx.x * blockDim.x + threadIdx.x;
  if (i >= NF) return;
  const int f = (int)(i % F);
  float v = out[i] + b[f];
  if (do_relu) v = fmaxf(v, 0.0f);
  out[i] = v;
}

// layer 3 fused GEMV (K x 1) + scalar attention coefficients
__global__ void gemv_alpha(const float* __restrict__ h,
                           const float* __restrict__ W3,
                           const float* __restrict__ as3,
                           const float* __restrict__ ad3,
                           float* __restrict__ h3, float* __restrict__ asrc,
                           float* __restrict__ adst, int N, int K) {
  int i = blockIdx.x * blockDim.x + threadIdx.x;
  if (i >= N) return;
  const float* row = h + (size_t)i * K;
  float acc = 0.0f;
  for (int k = 0; k < K; ++k) acc = fmaf(row[k], W3[k], acc);
  h3[i] = acc;
  asrc[i] = acc * as3[0];
  adst[i] = acc * ad3[0];
}

// ---------------------------------------------------------------------------
// Orchestration
// ---------------------------------------------------------------------------
static inline int cdiv_i(long long a, long long b) { return (int)((a + b - 1) / b); }

extern "C" void kernel_launch(void* const* d_in, const int* in_sizes, int n_in,
                              void* d_out, int out_size, void* d_ws, size_t ws_size,
                              hipStream_t stream) {
  const float* x = (const float*)d_in[0];
  const long long* ei = (const long long*)d_in[1];  // int64 [2, E]
  // d_in[2] = edge_attr : accepted but unused (reference ignores it)
  const float* W1 = (const float*)d_in[3];
  const float* as1 = (const float*)d_in[4];
  const float* ad1 = (const float*)d_in[5];
  const float* b1 = (const float*)d_in[6];
  const float* W2 = (const float*)d_in[7];
  const float* as2 = (const float*)d_in[8];
  const float* ad2 = (const float*)d_in[9];
  const float* b2 = (const float*)d_in[10];
  const float* W3 = (const float*)d_in[11];
  const float* as3 = (const float*)d_in[12];
  const float* ad3 = (const float*)d_in[13];
  const float* b3 = (const float*)d_in[14];

  const int F_in = 256, F1 = 48, F2 = 60;
  const int N = in_sizes[0] / F_in;        // 50000 (multiple of 16)
  const long long E = in_sizes[1] / 2;     // 1.6M
  const long long ET = E + N;              // + self loops

  // workspace partition (floats)
  float* ws = (float*)d_ws;
  size_t o = 0;
  float* h1 = ws + o;  o += (size_t)N * F1;
  float* o1 = ws + o;  o += (size_t)N * F1;
  float* h2 = ws + o;  o += (size_t)N * F2;
  float* o2 = ws + o;  o += (size_t)N * F2;
  float* asb = ws + o; o += (size_t)N;
  float* adb = ws + o; o += (size_t)N;
  float* mb = ws + o;  o += (size_t)N;
  float* sb = ws + o;  o += (size_t)N;
  float* h3 = ws + o;  o += (size_t)N;
  float* elog = ws + o; o += (size_t)ET;
  (void)ws_size; (void)n_in; (void)out_size;

  const int TB = 256;
  const int nbN = cdiv_i(N, TB);
  const int nbE = cdiv_i(ET, TB);
  const float NEG_INF = -INFINITY;

  // ------------------------- Layer 1 (F=48, NT=3) -------------------------
  {
    const int mtiles = N / 16;
    wmma_gemm_f32<256, 3><<<cdiv_i(mtiles, 4), 128, 0, stream>>>(x, W1, h1, N, F_in, F1);
    node_alpha<<<nbN, TB, 0, stream>>>(h1, as1, ad1, asb, adb, N, F1);
    fill_f32<<<nbN, TB, 0, stream>>>(mb, NEG_INF, N);
    fill_f32<<<nbN, TB, 0, stream>>>(sb, 0.0f, N);
    fill_f32<<<cdiv_i((long long)N * F1, TB), TB, 0, stream>>>(o1, 0.0f, (long long)N * F1);
    edge_logit_max<<<nbE, TB, 0, stream>>>(ei, E, ET, asb, adb, elog, mb);
    sanitize_max<<<nbN, TB, 0, stream>>>(mb, N);
    edge_exp_sum<<<nbE, TB, 0, stream>>>(ei, E, ET, mb, elog, sb);
    edge_aggregate<<<nbE, TB, 0, stream>>>(ei, E, ET, elog, sb, h1, o1, F1);
    bias_act<<<cdiv_i((long long)N * F1, TB), TB, 0, stream>>>(o1, b1, (long long)N * F1, F1, 1);
  }

  // ------------------------- Layer 2 (F=60, NT=4) -------------------------
  {
    const int mtiles = N / 16;
    wmma_gemm_f32<48, 4><<<cdiv_i(mtiles, 4), 128, 0, stream>>>(o1, W2, h2, N, F1, F2);
    node_alpha<<<nbN, TB, 0, stream>>>(h2, as2, ad2, asb, adb, N, F2);
    fill_f32<<<nbN, TB, 0, stream>>>(mb, NEG_INF, N);
    fill_f32<<<nbN, TB, 0, stream>>>(sb, 0.0f, N);
    fill_f32<<<cdiv_i((long long)N * F2, TB), TB, 0, stream>>>(o2, 0.0f, (long long)N * F2);
    edge_logit_max<<<nbE, TB, 0, stream>>>(ei, E, ET, asb, adb, elog, mb);
    sanitize_max<<<nbN, TB, 0, stream>>>(mb, N);
    edge_exp_sum<<<nbE, TB, 0, stream>>>(ei, E, ET, mb, elog, sb);
    edge_aggregate<<<nbE, TB, 0, stream>>>(ei, E, ET, elog, sb, h2, o2, F2);
    bias_act<<<cdiv_i((long long)N * F2, TB), TB, 0, stream>>>(o2, b2, (long long)N * F2, F2, 1);
  }

  // ------------------------- Layer 3 (F=1) --------------------------------
  {
    float* out = (float*)d_out;
    gemv_alpha<<<nbN, TB, 0, stream>>>(o2, W3, as3, ad3, h3, asb, adb, N, F2);
    fill_f32<<<nbN, TB, 0, stream>>>(mb, NEG_INF, N);
    fill_f32<<<nbN, TB, 0, stream>>>(sb, 0.0f, N);
    fill_f32<<<nbN, TB, 0, stream>>>(out, 0.0f, N);
    edge_logit_max<<<nbE, TB, 0, stream>>>(ei, E, ET, asb, adb, elog, mb);
    sanitize_max<<<nbN, TB, 0, stream>>>(mb, N);
    edge_exp_sum<<<nbE, TB, 0, stream>>>(ei, E, ET, mb, elog, sb);
    edge_aggregate<<<nbE, TB, 0, stream>>>(ei, E, ET, elog, sb, h3, out, 1);
    bias_act<<<nbN, TB, 0, stream>>>(out, b3, N, 1, 0);
  }
}